// QuesAnsModel_28621662061019
// MI455X (gfx1250) — compile-verified
//
#include <hip/hip_runtime.h>

// ---------------------------------------------------------------------------
// MemN2N-style QA model on MI455X (gfx1250).
//
// Shapes: V=128000, D=256, M=200, HOPS=3.
// Heavy stage: current_A = memory @ WA.T + tA, current_C = memory @ WC.T + tC
//   -> two [200 x 128000] @ [128000 x 256] fp32 GEMMs (26 GFLOP, ~364 MB).
// Memory-bound (23.3 TB/s HBM => ~27us floor for the ~630MB total), so we
// keep fp32 end-to-end and use V_WMMA_F32_16X16X4_F32 for the GEMMs.
// Split-K (80 chunks of 1600) + f32 global atomics gives one-pass HBM traffic
// for the weights; the 1.28MB memory chunk per K-split stays hot in 192MB L2.
// ---------------------------------------------------------------------------

#define V_DIM 128000
#define D_DIM 256
#define M_DIM 200
#define HOPS 3
#define NSPLIT 80
#define KCHUNK (V_DIM / NSPLIT)   // 1600, divisible by 16
#define NDTILE (D_DIM / 16)       // 16
#define NMTILE 13                 // ceil(200/16)

typedef __attribute__((ext_vector_type(2))) float v2f;
typedef __attribute__((ext_vector_type(8))) float v8f;

// ---------------------------------------------------------------------------
// init: current_A = tA, current_C = tC (re-done every call; atomics accumulate)
// ---------------------------------------------------------------------------
__global__ void init_copy(const float* __restrict__ tA, const float* __restrict__ tC,
                          float* __restrict__ curA, float* __restrict__ curC) {
    int i = blockIdx.x * blockDim.x + threadIdx.x;
    if (i < M_DIM * D_DIM) {
        curA[i] = tA[i];
        curC[i] = tC[i];
    }
}

// ---------------------------------------------------------------------------
// Generic wave-per-row dot: y[r] = dot(x[0..L), Wm[r, 0..L)), r < R.
// Used for ques_d = question @ WB.T (R=256, L=128000)
// and     out    = ques_d  @ Wout.T (R=128000, L=256).
// Lanes stride float4 -> 512B contiguous per wave per load. wave32 shfl reduce.
// ---------------------------------------------------------------------------
__global__ void dot_rows(const float* __restrict__ x, const float* __restrict__ Wm,
                         float* __restrict__ y, int R, int L) {
    int wid  = (blockIdx.x * blockDim.x + threadIdx.x) >> 5;
    int lane = threadIdx.x & 31;
    if (wid >= R) return;
    const float4* wrow = (const float4*)(Wm + (size_t)wid * L);
    const float4* x4   = (const float4*)x;
    int n4 = L >> 2;
    float s = 0.0f;
    for (int i = lane; i < n4; i += 32) {
        float4 w = wrow[i];
        float4 q = x4[i];
        s += w.x * q.x + w.y * q.y + w.z * q.z + w.w * q.w;
    }
    #pragma unroll
    for (int off = 16; off; off >>= 1) s += __shfl_xor(s, off, 32);
    if (lane == 0) y[wid] = s;
}

// ---------------------------------------------------------------------------
// Split-K fp32 WMMA GEMM: out += Amat[0..M) x Wmat[dtile-stripe].T over K-chunk.
// grid = (16 d-tiles, 80 k-splits, 2 {WA->curA, WC->curC}); block = 13 waves,
// wave w owns M-tile w (16 rows), accumulates a 16x16 f32 tile in 8 VGPRs.
//
// V_WMMA_F32_16X16X4_F32 fragment layout (ISA 7.12.2):
//   A (16x4):  lane l holds A[l&15, (l>>4)*2 + {0,1}]  -> one float2 per step
//   B (4x16):  lane l holds B[(l>>4)*2 + {0,1}, l&15]  == W[dbase+(l&15), k..]
//   C/D:       vgpr j, lane l -> (m = j + (l>>4)*8, n = l&15)
// Both operands load a float2 at (row = lane&15 of tile, col = k0+(lane>>4)*2).
// ---------------------------------------------------------------------------
__global__ __launch_bounds__(416) void gemm_splitk(
    const float* __restrict__ Amat,   // memory [M, V]
    const float* __restrict__ W0,     // WA [D, V]
    const float* __restrict__ W1,     // WC [D, V]
    float* __restrict__ out0,         // current_A [M, D]
    float* __restrict__ out1) {       // current_C [M, D]
    const int dtile  = blockIdx.x;                    // 0..15
    const int ksplit = blockIdx.y;                    // 0..NSPLIT-1
    const float* Wmat = blockIdx.z ? W1 : W0;
    float* outp       = blockIdx.z ? out1 : out0;

    const int wave  = threadIdx.x >> 5;               // 0..12 -> M tile
    const int lane  = threadIdx.x & 31;
    const int mbase = wave * 16;
    const int dbase = dtile * 16;
    const int k0    = ksplit * KCHUNK;

    const int half = lane >> 4;                       // K-pair select
    const int rl   = lane & 15;                       // row within tile
    int arow = mbase + rl;
    if (arow > M_DIM - 1) arow = M_DIM - 1;           // clamp; results discarded
    const int brow = dbase + rl;

    const float* aptr = Amat + (size_t)arow * V_DIM + k0 + half * 2;
    const float* bptr = Wmat + (size_t)brow * V_DIM + k0 + half * 2;

    v8f acc = {};
    for (int k = 0; k < KCHUNK; k += 16) {
        v2f a0 = *(const v2f*)(aptr + 0);
        v2f b0 = *(const v2f*)(bptr + 0);
        v2f a1 = *(const v2f*)(aptr + 4);
        v2f b1 = *(const v2f*)(bptr + 4);
        v2f a2 = *(const v2f*)(aptr + 8);
        v2f b2 = *(const v2f*)(bptr + 8);
        v2f a3 = *(const v2f*)(aptr + 12);
        v2f b3 = *(const v2f*)(bptr + 12);
        acc = __builtin_amdgcn_wmma_f32_16x16x4_f32(false, a0, false, b0, (short)0, acc, false, false);
        acc = __builtin_amdgcn_wmma_f32_16x16x4_f32(false, a1, false, b1, (short)0, acc, false, false);
        acc = __builtin_amdgcn_wmma_f32_16x16x4_f32(false, a2, false, b2, (short)0, acc, false, false);
        acc = __builtin_amdgcn_wmma_f32_16x16x4_f32(false, a3, false, b3, (short)0, acc, false, false);
        aptr += 16;
        bptr += 16;
    }

    // Accumulate 16x16 tile into output with f32 atomics (split-K partials).
    #pragma unroll
    for (int j = 0; j < 8; ++j) {
        int m = mbase + j + half * 8;
        if (m < M_DIM) {
            __hip_atomic_fetch_add(outp + m * D_DIM + dbase + rl, acc[j],
                                   __ATOMIC_RELAXED, __HIP_MEMORY_SCOPE_AGENT);
        }
    }
}

// ---------------------------------------------------------------------------
// 3 hops of softmax memory addressing. Tiny (0.6 MFLOP): one block, 8 waves.
//   scores = ques_d @ curA.T ; P = softmax(scores) ; ques_d += P @ curC
// ---------------------------------------------------------------------------
__global__ __launch_bounds__(256) void hops_kernel(const float* __restrict__ curA,
                                                   const float* __restrict__ curC,
                                                   float* __restrict__ qd) {
    __shared__ __align__(16) float q[D_DIM];
    __shared__ float sc[256];
    __shared__ float red[8];
    const int t = threadIdx.x, lane = t & 31, wave = t >> 5;

    q[t] = qd[t];
    __syncthreads();

    for (int hop = 0; hop < HOPS; ++hop) {
        // ---- scores[m] = dot(q, curA[m,:]) : wave per m, 25 rows per wave
        const float4* q4 = (const float4*)q;
        float4 qa = q4[lane];
        float4 qb = q4[lane + 32];
        for (int m = wave; m < M_DIM; m += 8) {
            const float4* row = (const float4*)(curA + m * D_DIM);
            float4 a = row[lane];
            float4 b = row[lane + 32];
            float s = a.x * qa.x + a.y * qa.y + a.z * qa.z + a.w * qa.w
                    + b.x * qb.x + b.y * qb.y + b.z * qb.z + b.w * qb.w;
            #pragma unroll
            for (int off = 16; off; off >>= 1) s += __shfl_xor(s, off, 32);
            if (lane == 0) sc[m] = s;
        }
        __syncthreads();

        // ---- softmax over sc[0..M)
        float v = (t < M_DIM) ? sc[t] : -3.402823466e+38f;
        float r = v;
        #pragma unroll
        for (int off = 16; off; off >>= 1) r = fmaxf(r, __shfl_xor(r, off, 32));
        if (lane == 0) red[wave] = r;
        __syncthreads();
        float mx = red[0];
        #pragma unroll
        for (int i = 1; i < 8; ++i) mx = fmaxf(mx, red[i]);
        float e = (t < M_DIM) ? __expf(v - mx) : 0.0f;
        float se = e;
        #pragma unroll
        for (int off = 16; off; off >>= 1) se += __shfl_xor(se, off, 32);
        __syncthreads();                       // red re-use hazard
        if (lane == 0) red[wave] = se;
        __syncthreads();
        float tot = red[0];
        #pragma unroll
        for (int i = 1; i < 8; ++i) tot += red[i];
        sc[t] = e / tot;                       // P[m] (0 for t >= M)
        __syncthreads();

        // ---- q[d] += sum_m P[m] * curC[m, d] : thread t = column d (coalesced)
        float accd = q[t];
        for (int m = 0; m < M_DIM; ++m) accd += sc[m] * curC[m * D_DIM + t];
        __syncthreads();
        q[t] = accd;
        __syncthreads();
    }
    qd[t] = q[t];
}

// ---------------------------------------------------------------------------
// launch
// ---------------------------------------------------------------------------
extern "C" void kernel_launch(void* const* d_in, const int* in_sizes, int n_in,
                              void* d_out, int out_size, void* d_ws, size_t ws_size,
                              hipStream_t stream) {
    const float* question = (const float*)d_in[0];   // [1, V]
    const float* memory   = (const float*)d_in[1];   // [M, V]
    const float* WA       = (const float*)d_in[2];   // [D, V]
    const float* WB       = (const float*)d_in[3];   // [D, V]
    const float* WC       = (const float*)d_in[4];   // [D, V]
    const float* Wout     = (const float*)d_in[5];   // [V, D]
    const float* tA       = (const float*)d_in[6];   // [M, D]
    const float* tC       = (const float*)d_in[7];   // [M, D]

    float* curA = (float*)d_ws;                      // [M, D]
    float* curC = curA + M_DIM * D_DIM;              // [M, D]
    float* qd   = curC + M_DIM * D_DIM;              // [D]
    float* outp = (float*)d_out;                     // [V]

    // 1) current_A/C = temporal bias (atomics accumulate on top)
    init_copy<<<(M_DIM * D_DIM + 255) / 256, 256, 0, stream>>>(tA, tC, curA, curC);

    // 2) ques_d = question @ WB.T  (wave per output d)
    dot_rows<<<(D_DIM + 7) / 8, 256, 0, stream>>>(question, WB, qd, D_DIM, V_DIM);

    // 3) split-K fp32 WMMA GEMMs: curA += memory@WA.T, curC += memory@WC.T
    gemm_splitk<<<dim3(NDTILE, NSPLIT, 2), 416, 0, stream>>>(memory, WA, WC, curA, curC);

    // 4) three softmax hops (single block)
    hops_kernel<<<1, 256, 0, stream>>>(curA, curC, qd);

    // 5) out = ques_d @ Wout.T  (wave per output v)
    dot_rows<<<(V_DIM + 7) / 8, 256, 0, stream>>>(qd, Wout, outp, V_DIM, D_DIM);
}